// Linear_9182640079252
// MI455X (gfx1250) — compile-verified
//
#include <hip/hip_runtime.h>
#include <hip/hip_bf16.h>

typedef float v2f __attribute__((ext_vector_type(2)));
typedef float v8f __attribute__((ext_vector_type(8)));

#define IN_F   4096
#define OUT_F  4096
#define NFAC   3
#define NBLK   64   // butterfly blocks per factor
#define BS     64   // block size

// ---------------------------------------------------------------------------
// 1) Cayley: Q = 2*(I+S)^-1 - I,  S = 0.5*(R - R^T).  One block per 64x64 mat.
//    Gauss-Jordan without pivoting (I+S has SPD symmetric part -> stable).
// ---------------------------------------------------------------------------
__global__ __launch_bounds__(256) void cayley_kernel(const float* __restrict__ R,
                                                     float* __restrict__ Q) {
    __shared__ float M[64][130];           // [A | I], padded stride
    const float* Rb = R + (size_t)blockIdx.x * (BS * BS);
    const int t = threadIdx.x;             // 256 threads
    const int row = t >> 2;                // 0..63
    const int cseg = (t & 3) * 32;         // 0,32,64,96

    for (int j = 0; j < 32; ++j) {
        int c = cseg + j;
        float v;
        if (c < 64) {
            float s = 0.5f * (Rb[row * 64 + c] - Rb[c * 64 + row]);
            v = (row == c ? 1.0f : 0.0f) + s;
        } else {
            v = ((c - 64) == row) ? 1.0f : 0.0f;
        }
        M[row][c] = v;
    }
    __syncthreads();

    for (int k = 0; k < 64; ++k) {
        float piv = M[k][k];               // read by all before any write
        __syncthreads();
        if (row == k) {
            float inv = 1.0f / piv;
            for (int j = 0; j < 32; ++j) M[k][cseg + j] *= inv;
        }
        __syncthreads();
        if (row != k) {
            float f = M[row][k];
            for (int j = 0; j < 32; ++j) {
                int c = cseg + j;
                M[row][c] -= f * M[k][c];
            }
        }
        __syncthreads();
    }

    float* Qb = Q + (size_t)blockIdx.x * (BS * BS);
    for (int j = 0; j < 32; ++j) {
        int c = cseg + j;
        if (c >= 64) {
            int cc = c - 64;
            Qb[row * 64 + cc] = 2.0f * M[row][c] - (row == cc ? 1.0f : 0.0f);
        }
    }
}

// ---------------------------------------------------------------------------
// 2) inverse permutations: ip[f][ perm[f][v] ] = v
// ---------------------------------------------------------------------------
__global__ void invperm_kernel(const int* __restrict__ perm, int* __restrict__ ip, int n) {
    int i = blockIdx.x * blockDim.x + threadIdx.x;
    if (i < n) {
        int f = i >> 12;
        int v = i & 4095;
        ip[f * IN_F + perm[i]] = v;
    }
}

// ---------------------------------------------------------------------------
// 3) rot0 dense materialization: rot[r][c] = Q0[blk][ar][bc] if blocks match
// ---------------------------------------------------------------------------
__global__ __launch_bounds__(256) void rot_init_kernel(const float* __restrict__ Q0,
                                                       const int* __restrict__ p0,
                                                       float* __restrict__ rot) {
    unsigned idx = blockIdx.x * 256u + threadIdx.x;   // 16M elements
    int r = idx >> 12;
    int c = idx & 4095;
    int a = p0[r];
    int b = p0[c];
    float v = 0.0f;
    if ((a >> 6) == (b >> 6))
        v = Q0[(size_t)(a >> 6) * 4096 + (a & 63) * 64 + (b & 63)];
    rot[idx] = v;
}

// ---------------------------------------------------------------------------
// 4) sparse butterfly apply: dst = (P B P^T) @ src
//    dst[ip[blk*64+ar]][c] = sum_j Q[blk][ar][j] * src[ip[blk*64+j]][c]
//    grid = (64 column tiles of 64, 64 blk)
// ---------------------------------------------------------------------------
__global__ __launch_bounds__(256) void rot_mul_kernel(const float* __restrict__ Qf,
                                                      const int* __restrict__ ipf,
                                                      const float* __restrict__ src,
                                                      float* __restrict__ dst) {
    __shared__ float Qs[64 * 64];
    __shared__ int rows[64];
    const int t = threadIdx.x;
    const float* Qb = Qf + (size_t)blockIdx.y * 4096;
    #pragma unroll
    for (int i = 0; i < 16; ++i) Qs[t + 256 * i] = Qb[t + 256 * i];
    if (t < 64) rows[t] = ipf[blockIdx.y * 64 + t];
    __syncthreads();

    const int col = blockIdx.x * 64 + (t & 63);
    const int g = t >> 6;                  // 0..3 -> handles ar = g*16 .. g*16+15
    float acc[16];
    #pragma unroll
    for (int i = 0; i < 16; ++i) acc[i] = 0.0f;

    for (int j = 0; j < 64; ++j) {
        float v = src[(size_t)rows[j] * 4096 + col];
        #pragma unroll
        for (int i = 0; i < 16; ++i)
            acc[i] += Qs[(g * 16 + i) * 64 + j] * v;   // uniform per wave -> LDS broadcast
    }
    #pragma unroll
    for (int i = 0; i < 16; ++i)
        dst[(size_t)rows[g * 16 + i] * 4096 + col] = acc[i];
}

// ---------------------------------------------------------------------------
// 5) NT GEMM with fp32 WMMA:  C[m][n] = (sum_k A[m,k]*B[n,k]) * rs[m]? + cb[n]?
//    128x128 tile / block (8 wave32 in 2x4), K-chunk 32 in LDS, pad stride 36.
// ---------------------------------------------------------------------------
#define GT_M 128
#define GT_N 128
#define GT_K 32
#define LDSS 36

__global__ __launch_bounds__(256) void gemm_nt_wmma(const float* __restrict__ A,
                                                    const float* __restrict__ B,
                                                    const float* __restrict__ row_scale,
                                                    const float* __restrict__ col_bias,
                                                    float* __restrict__ C,
                                                    int M, int N, int K) {
    __shared__ float As[GT_M * LDSS];
    __shared__ float Bs[GT_N * LDSS];

    const int t    = threadIdx.x;
    const int lane = t & 31;
    const int w    = t >> 5;       // 0..7
    const int wm   = w >> 2;       // 0..1  -> 64 rows of M
    const int wn   = w & 3;        // 0..3  -> 32 cols of N

    const int m0 = blockIdx.y * GT_M;
    const int n0 = blockIdx.x * GT_N;

    v8f vzero = {0.f, 0.f, 0.f, 0.f, 0.f, 0.f, 0.f, 0.f};
    v8f acc[4][2];
    #pragma unroll
    for (int mi = 0; mi < 4; ++mi)
        #pragma unroll
        for (int ni = 0; ni < 2; ++ni) acc[mi][ni] = vzero;

    const int lrow = t >> 3;            // 0..31
    const int lcol = (t & 7) * 4;       // 0,4,...,28
    const int kh = (lane >> 4) * 2;     // 0 or 2 : K pair select (A/B frag layout)
    const int ml = lane & 15;           // M (or N) index within 16

    for (int k0 = 0; k0 < K; k0 += GT_K) {
        #pragma unroll
        for (int rr = 0; rr < 4; ++rr) {
            int r = lrow + rr * 32;
            float4 av = *(const float4*)(A + (size_t)(m0 + r) * K + k0 + lcol);
            float4 bv = *(const float4*)(B + (size_t)(n0 + r) * K + k0 + lcol);
            *(float4*)(&As[r * LDSS + lcol]) = av;
            *(float4*)(&Bs[r * LDSS + lcol]) = bv;
        }
        __syncthreads();

        #pragma unroll
        for (int kk = 0; kk < GT_K; kk += 4) {
            v2f a_frag[4], b_frag[2];
            #pragma unroll
            for (int mi = 0; mi < 4; ++mi)
                a_frag[mi] = *(const v2f*)(&As[(wm * 64 + mi * 16 + ml) * LDSS + kk + kh]);
            #pragma unroll
            for (int ni = 0; ni < 2; ++ni)
                b_frag[ni] = *(const v2f*)(&Bs[(wn * 32 + ni * 16 + ml) * LDSS + kk + kh]);
            #pragma unroll
            for (int mi = 0; mi < 4; ++mi)
                #pragma unroll
                for (int ni = 0; ni < 2; ++ni)
                    acc[mi][ni] = __builtin_amdgcn_wmma_f32_16x16x4_f32(
                        false, a_frag[mi], false, b_frag[ni],
                        (short)0, acc[mi][ni], false, false);
        }
        __syncthreads();
    }

    // Epilogue: C/D layout -> VGPR v: M = v + 8*(lane>=16), N = lane&15
    const int nl = lane & 15;
    const int mh = (lane >> 4) * 8;
    #pragma unroll
    for (int ni = 0; ni < 2; ++ni) {
        int ng = n0 + wn * 32 + ni * 16 + nl;
        float cb = col_bias ? col_bias[ng] : 0.0f;
        #pragma unroll
        for (int mi = 0; mi < 4; ++mi) {
            int mbase = m0 + wm * 64 + mi * 16 + mh;
            #pragma unroll
            for (int v = 0; v < 8; ++v) {
                int mg = mbase + v;
                float val = acc[mi][ni][v];
                if (row_scale) val *= row_scale[mg];
                C[(size_t)mg * N + ng] = val + cb;
            }
        }
    }
}

// ---------------------------------------------------------------------------
// launch
// ---------------------------------------------------------------------------
extern "C" void kernel_launch(void* const* d_in, const int* in_sizes, int n_in,
                              void* d_out, int out_size, void* d_ws, size_t ws_size,
                              hipStream_t stream) {
    const float* x      = (const float*)d_in[0];   // (8,2048,4096)
    const float* weight = (const float*)d_in[1];   // (4096,4096)
    const float* bias   = (const float*)d_in[2];   // (4096)
    const float* boft_R = (const float*)d_in[3];   // (3,64,64,64)
    const float* boft_s = (const float*)d_in[4];   // (4096,1)
    const int*   perm   = (const int*)d_in[5];     // (3,4096)
    float* out = (float*)d_out;                    // (8,2048,4096)

    char* ws = (char*)d_ws;
    const size_t Q_BYTES  = (size_t)NFAC * NBLK * BS * BS * sizeof(float);   // 3,145,728
    const size_t IP_BYTES = (size_t)NFAC * IN_F * sizeof(int);               //    49,152
    const size_t MAT_BYTES = (size_t)IN_F * IN_F * sizeof(float);            // 67,108,864
    float* Qm   = (float*)ws;
    int*   ip   = (int*)(ws + Q_BYTES);
    float* bufA = (float*)(ws + Q_BYTES + IP_BYTES);
    float* bufB = (float*)(ws + Q_BYTES + IP_BYTES + MAT_BYTES);
    (void)ws_size; (void)in_sizes; (void)n_in; (void)out_size;

    // 1) Cayley on 192 blocks
    cayley_kernel<<<NFAC * NBLK, 256, 0, stream>>>(boft_R, Qm);
    // 2) inverse perms
    invperm_kernel<<<(NFAC * IN_F + 255) / 256, 256, 0, stream>>>(perm, ip, NFAC * IN_F);
    // 3) rot0 dense
    rot_init_kernel<<<(IN_F * IN_F) / 256, 256, 0, stream>>>(Qm, perm, bufA);
    // 4) rot1 = mat1 @ rot0 ; rot2 = mat2 @ rot1
    rot_mul_kernel<<<dim3(IN_F / 64, NBLK), 256, 0, stream>>>(
        Qm + (size_t)1 * NBLK * BS * BS, ip + 1 * IN_F, bufA, bufB);
    rot_mul_kernel<<<dim3(IN_F / 64, NBLK), 256, 0, stream>>>(
        Qm + (size_t)2 * NBLK * BS * BS, ip + 2 * IN_F, bufB, bufA);   // rot -> bufA
    // 5) scaled_w[o][i] = (sum_k W[o][k] * rot[i][k]) * s[o]   -> bufB
    gemm_nt_wmma<<<dim3(IN_F / GT_N, OUT_F / GT_M), 256, 0, stream>>>(
        weight, bufA, boft_s, nullptr, bufB, OUT_F, IN_F, IN_F);
    // 6) out[t][o] = sum_i x[t][i] * scaled_w[o][i] + bias[o]
    gemm_nt_wmma<<<dim3(OUT_F / GT_N, (8 * 2048) / GT_M), 256, 0, stream>>>(
        x, bufB, nullptr, bias, out, 8 * 2048, OUT_F, IN_F);
}